// SelfAttention_56178172231887
// MI455X (gfx1250) — compile-verified
//
#include <hip/hip_runtime.h>

// CDNA5 / gfx1250 fused self-attention (no-softmax variant, faithful to reference)
// Stage 1: O_pre[n,q,h*64+d] = sum_l where(mask,QK^T,-1e20)[l,q-tile] * V   (per head)
// Stage 2: out = O_pre @ W^T + b
// fp32 math via V_WMMA_F32_16X16X4_F32; tiles staged with double-buffered
// GLOBAL_LOAD_ASYNC_TO_LDS_B128 (ASYNCcnt) so global->LDS overlaps the WMMA chain.

typedef float v2f __attribute__((ext_vector_type(2)));
typedef float v8f __attribute__((ext_vector_type(8)));
typedef unsigned int u32;

#define S_LEN   2048
#define EMB     512
#define NBATCH  4
#define NHEAD   8
#define HDIM    64
#define KSTRIDE 516   // 512 + 4 pad: stride%64==4 -> conflict-free lane patterns
#define ASTRIDE 36    // 32 + 4 pad for stage-2 tiles

static __device__ __forceinline__ v8f wmma_f32(v2f a, v2f b, v8f c) {
  return __builtin_amdgcn_wmma_f32_16x16x4_f32(false, a, false, b, (short)0, c,
                                               false, false);
}

// One async 16B global->LDS copy; INST_OFFSET applies to both addresses.
#define ASYNC_B128(ldsa, voff, sbase, IMM)                                   \
  asm volatile("global_load_async_to_lds_b128 %0, %1, %2 offset:" #IMM       \
               :: "v"(ldsa), "v"(voff), "s"(sbase) : "memory")

// Copy a contiguous 128-byte per-thread segment (8 x b128).
static __device__ __forceinline__ void async_copy_128B(u32 ldsa, u32 voff,
                                                       const float* sbase) {
  ASYNC_B128(ldsa, voff, sbase, 0);
  ASYNC_B128(ldsa, voff, sbase, 16);
  ASYNC_B128(ldsa, voff, sbase, 32);
  ASYNC_B128(ldsa, voff, sbase, 48);
  ASYNC_B128(ldsa, voff, sbase, 64);
  ASYNC_B128(ldsa, voff, sbase, 80);
  ASYNC_B128(ldsa, voff, sbase, 96);
  ASYNC_B128(ldsa, voff, sbase, 112);
}

// Copy a contiguous 64-byte per-thread segment (4 x b128).
static __device__ __forceinline__ void async_copy_64B(u32 ldsa, u32 voff,
                                                      const float* sbase) {
  ASYNC_B128(ldsa, voff, sbase, 0);
  ASYNC_B128(ldsa, voff, sbase, 16);
  ASYNC_B128(ldsa, voff, sbase, 32);
  ASYNC_B128(ldsa, voff, sbase, 48);
}

static __device__ __forceinline__ u32 lds_addr(const void* p) {
  // __shared__ generic pointer: low 32 bits = workgroup-relative LDS byte addr
  return (u32)(unsigned long long)p;
}

// ---------------------------------------------------------------------------
// Kernel 1: fused masked-scores x V.
// grid = (S/32, N), block = 256 (8 waves). wave w = head w, q-rows [q0,q0+32).
// Per wave: S'[l,q] = K·Q^T tiles (M=l on VGPR rows, N=q on lanes), mask,
// half-swap shuffle to reuse S' as B operand of O'[d,q] = V^T · S'.
// ---------------------------------------------------------------------------
__global__ __launch_bounds__(256)
void attn_fused_kernel(const float* __restrict__ key,
                       const float* __restrict__ query,
                       const float* __restrict__ value,
                       const int*   __restrict__ mask,
                       float*       __restrict__ opre)
{
  __shared__ float lds_k[2][16 * KSTRIDE];
  __shared__ float lds_v[2][16 * KSTRIDE];

  const int n    = blockIdx.y;
  const int q0   = blockIdx.x * 32;
  const int tid  = threadIdx.x;
  const int h    = tid >> 5;        // wave id == head
  const int lane = tid & 31;
  const int l16  = lane & 15;
  const int hi   = lane >> 4;       // 0: lanes 0-15, 1: lanes 16-31

  // Async-staging geometry: thread -> 128B of one row (16 l-rows x 512 dims).
  const int srow = tid >> 4;           // 0..15
  const int scg  = (tid & 15) * 32;    // 32 floats
  const u32 voff = (u32)((srow * EMB + scg) * 4);
  u32 ldsK[2], ldsV[2];
  ldsK[0] = lds_addr(&lds_k[0][srow * KSTRIDE + scg]);
  ldsK[1] = lds_addr(&lds_k[1][srow * KSTRIDE + scg]);
  ldsV[0] = lds_addr(&lds_v[0][srow * KSTRIDE + scg]);
  ldsV[1] = lds_addr(&lds_v[1][srow * KSTRIDE + scg]);
  const float* kbase = key   + (size_t)n * S_LEN * EMB;
  const float* vbase = value + (size_t)n * S_LEN * EMB;

  // Prologue: tile 0 -> buffer 0 (16 async b128 per thread in flight).
  async_copy_128B(ldsK[0], voff, kbase);
  async_copy_128B(ldsV[0], voff, vbase);

  // Preload Q^T as B-operand: lane -> q = q0 + qs*16 + l16 (N dim),
  // chunk c covers k=d in {4c+2*hi, 4c+2*hi+1}.
  v2f qb[2][16];
#pragma unroll
  for (int qs = 0; qs < 2; ++qs) {
    const float* qrow =
        query + (size_t)(n * S_LEN + q0 + qs * 16 + l16) * EMB + h * HDIM;
#pragma unroll
    for (int c = 0; c < 16; ++c)
      qb[qs][c] = *(const v2f*)(qrow + 4 * c + 2 * hi);
  }

  v8f acc[2][4];
#pragma unroll
  for (int qs = 0; qs < 2; ++qs)
#pragma unroll
    for (int dt = 0; dt < 4; ++dt)
      acc[qs][dt] = (v8f){0.f, 0.f, 0.f, 0.f, 0.f, 0.f, 0.f, 0.f};

  const int* mbase = mask + (size_t)n * S_LEN * S_LEN;

  for (int it = 0; it < S_LEN / 16; ++it) {
    const int l0 = it * 16;
    // Issue next tile into the other buffer; prior-iteration trailing barrier
    // guarantees every wave is done reading it.
    if (it + 1 < S_LEN / 16) {
      const int nb = (it + 1) & 1;
      const size_t nxt = (size_t)(it + 1) * 16 * EMB;
      async_copy_128B(ldsK[nb], voff, kbase + nxt);
      async_copy_128B(ldsV[nb], voff, vbase + nxt);
      asm volatile("s_wait_asynccnt 0x10" ::: "memory");  // tile `it` landed
    } else {
      asm volatile("s_wait_asynccnt 0x0" ::: "memory");
    }
    __syncthreads();  // tile `it` visible to all waves

    const int cb = it & 1;
    const float* lk = &lds_k[cb][0];
    const float* lv = &lds_v[cb][0];

    // A operand of GEMM1: K strip, lane -> l = l16 (M dim), k=d chunks.
    v2f kb[16];
#pragma unroll
    for (int c = 0; c < 16; ++c)
      kb[c] = *(const v2f*)(lk + l16 * KSTRIDE + h * HDIM + 4 * c + 2 * hi);

    v2f sbq[2][4];
#pragma unroll
    for (int qs = 0; qs < 2; ++qs) {
      // S'[l,q] over k=d (64) = 16 chained K=4 WMMAs.
      v8f s = (v8f){0.f, 0.f, 0.f, 0.f, 0.f, 0.f, 0.f, 0.f};
#pragma unroll
      for (int c = 0; c < 16; ++c)
        s = wmma_f32(kb[c], qb[qs][c], s);

      // Mask: element (l = r+8*hi, q = l16). Per lane: 8 consecutive l.
      {
        const int* mrow = mbase + (size_t)(q0 + qs * 16 + l16) * S_LEN + l0 + 8 * hi;
        int4 ma = *(const int4*)(mrow);
        int4 mb = *(const int4*)(mrow + 4);
        s[0] = ma.x ? s[0] : -1e20f;
        s[1] = ma.y ? s[1] : -1e20f;
        s[2] = ma.z ? s[2] : -1e20f;
        s[3] = ma.w ? s[3] : -1e20f;
        s[4] = mb.x ? s[4] : -1e20f;
        s[5] = mb.y ? s[5] : -1e20f;
        s[6] = mb.z ? s[6] : -1e20f;
        s[7] = mb.w ? s[7] : -1e20f;
      }

      // C/D layout -> B layout: rows 0-7 live in lanes 0-15, rows 8-15 in
      // lanes 16-31; B chunks need the opposite half for half the lanes.
      float xs[8];
#pragma unroll
      for (int r = 0; r < 8; ++r) xs[r] = __shfl_xor(s[r], 16, 32);
      sbq[qs][0] = hi ? (v2f){xs[2], xs[3]} : (v2f){s[0], s[1]};
      sbq[qs][1] = hi ? (v2f){xs[6], xs[7]} : (v2f){s[4], s[5]};
      sbq[qs][2] = hi ? (v2f){s[2], s[3]}   : (v2f){xs[0], xs[1]};
      sbq[qs][3] = hi ? (v2f){s[6], s[7]}   : (v2f){xs[4], xs[5]};
    }

    // GEMM2: O'[d,q] += V^T · S'. A = V^T: lane -> d (M), k = l chunks.
#pragma unroll
    for (int dt = 0; dt < 4; ++dt) {
#pragma unroll
      for (int cc = 0; cc < 4; ++cc) {
        const float* vp =
            lv + (4 * cc + 2 * hi) * KSTRIDE + h * HDIM + dt * 16 + l16;
        v2f va = {vp[0], vp[KSTRIDE]};
        acc[0][dt] = wmma_f32(va, sbq[0][cc], acc[0][dt]);
        acc[1][dt] = wmma_f32(va, sbq[1][cc], acc[1][dt]);
      }
    }
    __syncthreads();  // all waves done with buffer cb before it is re-filled
  }

  // Store O' (C/D layout: N=q on lanes, M=d on vgprs -> 8 consecutive d per
  // lane per tile -> two float4 stores each).
#pragma unroll
  for (int qs = 0; qs < 2; ++qs) {
#pragma unroll
    for (int dt = 0; dt < 4; ++dt) {
      float* op = opre + (size_t)(n * S_LEN + q0 + qs * 16 + l16) * EMB +
                  h * HDIM + dt * 16 + 8 * hi;
      v8f a = acc[qs][dt];
      *(float4*)(op) = make_float4(a[0], a[1], a[2], a[3]);
      *(float4*)(op + 4) = make_float4(a[4], a[5], a[6], a[7]);
    }
  }
}

// ---------------------------------------------------------------------------
// Kernel 2: out = O_pre @ W^T + b.  Block tile 128r x 128j, wave w -> 16 rows.
// Same double-buffered async-to-LDS pipeline over 32-wide k slabs.
// ---------------------------------------------------------------------------
__global__ __launch_bounds__(256)
void linear_out_kernel(const float* __restrict__ opre,
                       const float* __restrict__ W,
                       const float* __restrict__ bias,
                       float*       __restrict__ out)
{
  __shared__ float lds_a[2][128 * ASTRIDE];
  __shared__ float lds_w[2][128 * ASTRIDE];

  const int tid  = threadIdx.x;
  const int wave = tid >> 5;
  const int lane = tid & 31;
  const int l16  = lane & 15;
  const int hi   = lane >> 4;
  const int r0   = blockIdx.x * 128;
  const int j0   = blockIdx.y * 128;

  // Async-staging geometry: thread -> 64B of one row (128 rows x 32 cols).
  const int srow = tid >> 1;          // 0..127
  const int scg  = (tid & 1) * 16;    // 16 floats
  const u32 voff = (u32)((srow * EMB + scg) * 4);
  u32 ldsA[2], ldsW[2];
  ldsA[0] = lds_addr(&lds_a[0][srow * ASTRIDE + scg]);
  ldsA[1] = lds_addr(&lds_a[1][srow * ASTRIDE + scg]);
  ldsW[0] = lds_addr(&lds_w[0][srow * ASTRIDE + scg]);
  ldsW[1] = lds_addr(&lds_w[1][srow * ASTRIDE + scg]);
  const float* abase = opre + (size_t)r0 * EMB;
  const float* wbase = W    + (size_t)j0 * EMB;

  // Prologue: k-slab 0 -> buffer 0 (8 async b128 per thread in flight).
  async_copy_64B(ldsA[0], voff, abase);
  async_copy_64B(ldsW[0], voff, wbase);

  v8f acc[8];
#pragma unroll
  for (int jt = 0; jt < 8; ++jt) {
    float bj = bias[j0 + jt * 16 + l16];
    acc[jt] = (v8f){bj, bj, bj, bj, bj, bj, bj, bj};
  }

  for (int it = 0; it < EMB / 32; ++it) {
    if (it + 1 < EMB / 32) {
      const int nb = (it + 1) & 1;
      async_copy_64B(ldsA[nb], voff, abase + (it + 1) * 32);
      async_copy_64B(ldsW[nb], voff, wbase + (it + 1) * 32);
      asm volatile("s_wait_asynccnt 0x8" ::: "memory");
    } else {
      asm volatile("s_wait_asynccnt 0x0" ::: "memory");
    }
    __syncthreads();

    const int cb = it & 1;
    // A = O_pre: lane -> r (M); B = W^T: lane -> j (N); k = e chunks.
    v2f av[8];
#pragma unroll
    for (int c = 0; c < 8; ++c)
      av[c] = *(const v2f*)(&lds_a[cb][(wave * 16 + l16) * ASTRIDE + 4 * c + 2 * hi]);
#pragma unroll
    for (int jt = 0; jt < 8; ++jt) {
#pragma unroll
      for (int c = 0; c < 8; ++c) {
        v2f bw = *(const v2f*)(&lds_w[cb][(jt * 16 + l16) * ASTRIDE + 4 * c + 2 * hi]);
        acc[jt] = wmma_f32(av[c], bw, acc[jt]);
      }
    }
    __syncthreads();
  }

  // C/D layout: N=j on lanes, M=r on vgprs.
#pragma unroll
  for (int jt = 0; jt < 8; ++jt)
#pragma unroll
    for (int v = 0; v < 8; ++v)
      out[(size_t)(r0 + wave * 16 + v + 8 * hi) * EMB + j0 + jt * 16 + l16] =
          acc[jt][v];
}

extern "C" void kernel_launch(void* const* d_in, const int* in_sizes, int n_in,
                              void* d_out, int out_size, void* d_ws, size_t ws_size,
                              hipStream_t stream) {
  (void)in_sizes; (void)n_in; (void)out_size; (void)ws_size;
  const float* key   = (const float*)d_in[0];
  const float* query = (const float*)d_in[1];
  const float* value = (const float*)d_in[2];
  const int*   mask  = (const int*)d_in[3];
  const float* W     = (const float*)d_in[4];
  const float* bias  = (const float*)d_in[5];
  float* out  = (float*)d_out;
  float* opre = (float*)d_ws;   // N*S*EMB fp32 = 16.8 MB scratch

  dim3 g1(S_LEN / 32, NBATCH);
  attn_fused_kernel<<<g1, 256, 0, stream>>>(key, query, value, mask, opre);

  dim3 g2((NBATCH * S_LEN) / 128, EMB / 128);
  linear_out_kernel<<<g2, 256, 0, stream>>>(opre, W, bias, out);
}